// GIG_Transformer_27582279975007
// MI455X (gfx1250) — compile-verified
//
#include <hip/hip_runtime.h>
#include <math.h>

#define D_IN 128
#define NH   4
#define NC   32

typedef __attribute__((ext_vector_type(2))) float v2f;
typedef __attribute__((ext_vector_type(8))) float v8f;

// ---------------------------------------------------------------------------
// float atomic max via integer atomics (GLOBAL_ATOMIC_MAX_I32 / MIN_U32):
// IEEE floats order like ints when >=0 and like reversed unsigned ints when
// negative. amax initialized to -FLT_MAX.
__device__ __forceinline__ void atomicMaxF32(float* addr, float val) {
  if (val >= 0.0f) {
    atomicMax((int*)addr, __float_as_int(val));
  } else {
    atomicMin((unsigned int*)addr, (unsigned int)__float_as_int(val));
  }
}

// ---------------------------------------------------------------------------
// Init accumulators: accum[n*128]=0, amax[n*4]=-FLT_MAX, denom[n*4]=0
__global__ void init_kernel(float* __restrict__ amax, float* __restrict__ denom,
                            float* __restrict__ accum, int n) {
  int idx = blockIdx.x * blockDim.x + threadIdx.x;
  if (idx < n * D_IN) accum[idx] = 0.0f;
  if (idx < n * NH) {
    amax[idx]  = -3.402823e38f;
    denom[idx] = 0.0f;
  }
}

// ---------------------------------------------------------------------------
// Fused QKVS GEMM: one wave32 computes a 16x16 tile of the [N, 512] output
// (columns 0-127 -> q, 128-255 -> k, 256-383 -> v, 384-511 -> skip/d_out).
// V_WMMA_F32_16X16X4_F32 over K=128 (32 WMMAs). Bias folded into C init.
// grid = (N/16, 4); the 8 waves of a block share one X row-tile (WGP$ reuse).
__global__ __launch_bounds__(256) void qkvs_gemm(
    const float* __restrict__ x,
    const float* __restrict__ Wq, const float* __restrict__ bq,
    const float* __restrict__ Wk, const float* __restrict__ bk,
    const float* __restrict__ Wv, const float* __restrict__ bv,
    const float* __restrict__ Ws, const float* __restrict__ bs,
    float* __restrict__ q, float* __restrict__ k,
    float* __restrict__ v, float* __restrict__ skip, int n) {
  int lane = threadIdx.x & 31;
  int cb   = blockIdx.y * 8 + (threadIdx.x >> 5);  // column block 0..31 of 512
  int row0 = blockIdx.x * 16;

  int sel  = cb >> 3;          // which weight matrix
  int col0 = (cb & 7) * 16;    // column within that matrix

  const float* W; const float* bias; float* dst;
  switch (sel) {
    case 0:  W = Wq; bias = bq; dst = q;    break;
    case 1:  W = Wk; bias = bk; dst = k;    break;
    case 2:  W = Wv; bias = bv; dst = v;    break;
    default: W = Ws; bias = bs; dst = skip; break;
  }

  int m  = lane & 15;            // M index for A, N index for B
  int hk = (lane >> 4) << 1;     // 0 (lanes 0-15) or 2 (lanes 16-31)
  int arow = row0 + m; if (arow >= n) arow = n - 1;   // clamped read (tail safe)
  const float* xrow = x + (size_t)arow * D_IN;

  float bval = bias[col0 + m];
  v8f acc;
#pragma unroll
  for (int i = 0; i < 8; ++i) acc[i] = bval;

#pragma unroll 4
  for (int kk = 0; kk < D_IN; kk += 4) {
    v2f A, B;
    // f32 A 16x4 layout: VGPR0 = K{0,2}, VGPR1 = K{1,3} split across half-waves
    A.x = xrow[kk + hk];
    A.y = xrow[kk + hk + 1];
    // f32 B 4x16 layout mirrors A: rows striped across lanes per VGPR
    B.x = W[(kk + hk) * D_IN + col0 + m];
    B.y = W[(kk + hk + 1) * D_IN + col0 + m];
    acc = __builtin_amdgcn_wmma_f32_16x16x4_f32(
        /*neg_a=*/false, A, /*neg_b=*/false, B,
        /*c_mod=*/(short)0, acc, /*reuse_a=*/false, /*reuse_b=*/false);
  }

  // C/D layout: VGPR i holds (M = i + 8*(lane>=16), N = lane&15)
  int rbase = row0 + ((lane >> 4) << 3);
  float* dp = dst + (size_t)rbase * D_IN + col0 + m;
  if (row0 + 16 <= n) {          // full tile: unguarded store burst
#pragma unroll
    for (int i = 0; i < 8; ++i) dp[(size_t)i * D_IN] = acc[i];
  } else {                       // ragged tail (not hit when N%16==0)
#pragma unroll
    for (int i = 0; i < 8; ++i)
      if (rbase + i < n) dp[(size_t)i * D_IN] = acc[i];
  }
}

// ---------------------------------------------------------------------------
// Edge pass A: per (edge, head) logit = <q[dst,h,:], k[src,h,:]>/sqrt(C),
// store logit, atomic-max into amax[dst,h].
__global__ void edge_logits(const int* __restrict__ ei, int E,
                            const float* __restrict__ q,
                            const float* __restrict__ k,
                            float* __restrict__ logits,
                            float* __restrict__ amax) {
  int idx = blockIdx.x * blockDim.x + threadIdx.x;
  if (idx >= E * NH) return;
  int e = idx >> 2, h = idx & 3;
  int src = ei[e];
  int dst = ei[E + e];
  const float4* qp = (const float4*)(q + (size_t)dst * D_IN + h * NC);
  const float4* kp = (const float4*)(k + (size_t)src * D_IN + h * NC);
  float acc = 0.0f;
#pragma unroll
  for (int i = 0; i < 8; ++i) {
    float4 a = qp[i], b = kp[i];
    acc += a.x * b.x + a.y * b.y + a.z * b.z + a.w * b.w;
  }
  float lg = acc * 0.17677669529663689f;   // 1/sqrt(32)
  logits[idx] = lg;
  atomicMaxF32(&amax[dst * NH + h], lg);
}

// ---------------------------------------------------------------------------
// Edge pass B1: w = exp(logit - amax[dst,h]); overwrite logits with w
// (each thread touches only its own element -> race-free); denom += w.
__global__ void edge_exp(const int* __restrict__ ei, int E,
                         float* __restrict__ wbuf,        // in: logits, out: w
                         const float* __restrict__ amax,
                         float* __restrict__ denom) {
  int idx = blockIdx.x * blockDim.x + threadIdx.x;
  if (idx >= E * NH) return;
  int e = idx >> 2, h = idx & 3;
  int dst = ei[E + e];
  float w = __expf(wbuf[idx] - amax[dst * NH + h]);
  wbuf[idx] = w;
  atomicAdd(&denom[dst * NH + h], w);
}

// ---------------------------------------------------------------------------
// Edge pass B2: thread per (edge, h, c). A wave's 32 lanes cover 32
// contiguous channels of one edge -> coalesced v load and a single-cacheline
// atomic burst into accum[dst].
__global__ void edge_scatter(const int* __restrict__ ei, int E, int total,
                             const float* __restrict__ v,
                             const float* __restrict__ wbuf,
                             float* __restrict__ accum) {
  int idx = blockIdx.x * blockDim.x + threadIdx.x;
  if (idx >= total) return;
  int e = idx >> 7;            // edge
  int j = idx & 127;           // h*32 + c
  int h = j >> 5;
  int src = ei[e];
  int dst = ei[E + e];
  float w = wbuf[e * NH + h];
  atomicAdd(&accum[(size_t)dst * D_IN + j], w * v[(size_t)src * D_IN + j]);
}

// ---------------------------------------------------------------------------
// Final: out (already = skip) += accum / (denom + 1e-16)
__global__ void finalize(const float* __restrict__ accum,
                         const float* __restrict__ denom,
                         float* __restrict__ out, int n) {
  int idx = blockIdx.x * blockDim.x + threadIdx.x;
  if (idx >= n * D_IN) return;
  int node = idx >> 7;
  int h    = (idx >> 5) & 3;
  out[idx] += accum[idx] / (denom[node * NH + h] + 1e-16f);
}

// ---------------------------------------------------------------------------
extern "C" void kernel_launch(void* const* d_in, const int* in_sizes, int n_in,
                              void* d_out, int out_size, void* d_ws, size_t ws_size,
                              hipStream_t stream) {
  const float* x  = (const float*)d_in[0];
  const int*   ei = (const int*)d_in[1];
  const float* Wq = (const float*)d_in[2]; const float* bq = (const float*)d_in[3];
  const float* Wk = (const float*)d_in[4]; const float* bk = (const float*)d_in[5];
  const float* Wv = (const float*)d_in[6]; const float* bv = (const float*)d_in[7];
  const float* Ws = (const float*)d_in[8]; const float* bs = (const float*)d_in[9];

  int n = in_sizes[0] / D_IN;    // 50000
  int E = in_sizes[1] / 2;       // 800000

  float* q      = (float*)d_ws;
  float* k      = q     + (size_t)n * D_IN;
  float* v      = k     + (size_t)n * D_IN;
  float* accum  = v     + (size_t)n * D_IN;
  float* amax   = accum + (size_t)n * D_IN;
  float* denom  = amax  + (size_t)n * NH;
  float* logits = denom + (size_t)n * NH;
  float* out    = (float*)d_out;

  // 0) init accumulators
  init_kernel<<<(n * D_IN + 255) / 256, 256, 0, stream>>>(amax, denom, accum, n);

  // 1) fused QKV+skip GEMM (skip written straight into d_out)
  int n_row_tiles = (n + 15) / 16;                 // 3125
  qkvs_gemm<<<dim3(n_row_tiles, 4), 256, 0, stream>>>(
      x, Wq, bq, Wk, bk, Wv, bv, Ws, bs, q, k, v, out, n);

  // 2) per-edge logits + segment max
  int eh_blocks = (E * NH + 255) / 256;
  edge_logits<<<eh_blocks, 256, 0, stream>>>(ei, E, q, k, logits, amax);

  // 3) exp + denom (w overwrites logits in place)
  edge_exp<<<eh_blocks, 256, 0, stream>>>(ei, E, logits, amax, denom);

  // 4) coalesced weighted-V scatter
  int total = E * D_IN;                            // 102.4M threads
  edge_scatter<<<(total + 255) / 256, 256, 0, stream>>>(ei, E, total, v, logits,
                                                        accum);

  // 5) normalize + skip
  finalize<<<(n * D_IN + 255) / 256, 256, 0, stream>>>(accum, denom, out, n);
}